// EMStats_91259465105718
// MI455X (gfx1250) — compile-verified
//
#include <hip/hip_runtime.h>
#include <hip/hip_bf16.h>

typedef __attribute__((ext_vector_type(2))) float v2f;
typedef __attribute__((ext_vector_type(8))) float v8f;

#define L_DIM   8192
#define BD      128        // B*D
#define E_DIM   16
#define EBD     2048       // BD*E  (stride per timestep in (L,B,D,E) layout)
#define NSEG    16
#define SEGLEN  512        // L / NSEG
#define NCHUNK  32         // SEGLEN / 16
#define WARM    256        // MAX_T/2
#define EPS     1e-4f
#define PLANE   16777216   // L*BD*E
#define PAD     20         // LDS row stride (80B: keeps float4 stores 16B-aligned)

// workspace layout (float offsets)
#define WS_ALPHA       0
#define WS_OMA         16
#define WS_EMA0        32
#define WS_EMV0        160
#define WS_EMA_SEGEND  288
#define WS_EMA_CARRY   (288 + 32768)
#define WS_EMV_SEGEND  (288 + 2*32768)
#define WS_EMV_CARRY   (288 + 3*32768)   // total ~525 KB of ws

__device__ __forceinline__ v8f wmma4(v2f a, v2f b, v8f c) {
    // D(16x16,f32) += A(16x4,f32) x B(4x16,f32)
    return __builtin_amdgcn_wmma_f32_16x16x4_f32(
        false, a, false, b, (short)0, c, false, false);
}

__device__ __forceinline__ float powi(float b, int n) {
    float r = 1.f;
    for (int i = 0; i < n; ++i) r *= b;
    return r;
}

// A = lower-triangular scan matrix M[t][s] = (s<=t) ? alpha*oma^(t-s) : 0
// split into four 16x4 K-slices; ac = carry injector column oma^(t+1).
__device__ __forceinline__ void build_afrags(float alpha, float oma, int t, int h,
                                             v2f a[4], v2f& ac) {
#pragma unroll
    for (int k = 0; k < 4; ++k) {
        int s0 = 4 * k + 2 * h;                 // lanes 0-15: K = s0,s0+1 ; 16-31: +2
        a[k].x = (s0     <= t) ? alpha * powi(oma, t - s0)     : 0.f;
        a[k].y = (s0 + 1 <= t) ? alpha * powi(oma, t - s0 - 1) : 0.f;
    }
    ac.x = (h == 0) ? powi(oma, t + 1) : 0.f;   // B-row 0 pairs with carry row
    ac.y = 0.f;
}

__device__ __forceinline__ v8f scan_chunk(const v2f a[4], v2f ac,
                                          const v2f b[4], float carry, int h) {
    v2f bc;
    bc.x = (h == 0) ? carry : 0.f;  // row 0 of carry matrix = carry[c]
    bc.y = 0.f;
    v8f d = {0.f, 0.f, 0.f, 0.f, 0.f, 0.f, 0.f, 0.f};
    d = wmma4(a[0], b[0], d);
    d = wmma4(a[1], b[1], d);
    d = wmma4(a[2], b[2], d);
    d = wmma4(a[3], b[3], d);
    d = wmma4(ac, bc, d);           // + oma^(t+1) * carry  (rank-1 via WMMA)
    return d;
}

// Stage a 16x16 x-chunk into LDS with B128 loads/stores (each row = 64B).
__device__ __forceinline__ void stage_x(const float* __restrict__ x, int t0,
                                        int col0, int lane, float xs[16][PAD]) {
#pragma unroll
    for (int i = 0; i < 2; ++i) {
        int idx = i * 32 + lane;        // 0..63 float4 groups
        int tt = idx >> 2, cg = idx & 3;
        float4 v = *(const float4*)(x + (size_t)(t0 + tt) * BD + col0 + cg * 4);
        *(float4*)&xs[tt][cg * 4] = v;
    }
}

// ---------------- K1: alpha/oma + warm mean/var ----------------
__global__ __launch_bounds__(128)
void k_init(const float* __restrict__ x, const float* __restrict__ logd,
            float* __restrict__ ws) {
    int tid = threadIdx.x;
    if (tid < E_DIM) {
        float s = 1.f / (1.f + __expf(-logd[tid]));
        float decay = s * 510.f + 2.f;          // (MAX_T-MIN_T)+MIN_T
        float alpha = 2.f / (decay + 1.f);
        ws[WS_ALPHA + tid] = alpha;
        ws[WS_OMA + tid]   = 1.f - alpha;
    }
    float s = 0.f, s2 = 0.f;
    for (int t = 0; t < WARM; ++t) {
        float v = x[t * BD + tid];
        s += v; s2 += v * v;
    }
    float mean = s * (1.f / WARM);
    ws[WS_EMA0 + tid] = mean;
    ws[WS_EMV0 + tid] = s2 * (1.f / WARM) - mean * mean;
}

// ---------------- K2: ema local (zero-init) scan, keep segment ends ----------------
__global__ __launch_bounds__(32)
void k_ema_local(const float* __restrict__ x, float* __restrict__ ws) {
    __shared__ __align__(16) float xs[16][PAD];
    const int lane = threadIdx.x, c = lane & 15, h = lane >> 4;
    const int tile = blockIdx.x, seg = blockIdx.y, e = blockIdx.z;
    const int col0 = tile * 16;
    const float alpha = ws[WS_ALPHA + e], oma = ws[WS_OMA + e];
    v2f a[4], ac;
    build_afrags(alpha, oma, c, h, a, ac);

    float carry = 0.f;
    int t0 = seg * SEGLEN;
    for (int ch = 0; ch < NCHUNK; ++ch, t0 += 16) {
        stage_x(x, t0, col0, lane, xs);
        // prefetch next chunk: one 64B row per lane (rows are 512B apart)
        __builtin_prefetch(x + (size_t)(t0 + 16 + (lane & 15)) * BD + col0, 0, 3);
        __syncthreads();
        v2f b[4];
#pragma unroll
        for (int k = 0; k < 4; ++k) {
            int s0 = 4 * k + 2 * h;
            b[k].x = xs[s0][c];
            b[k].y = xs[s0 + 1][c];
        }
        v8f d = scan_chunk(a, ac, b, carry, h);
        carry = __shfl(d[7], 16 + c, 32);     // global t=15 row lives in lanes 16-31, reg 7
        __syncthreads();
    }
    if (lane < 16)
        ws[WS_EMA_SEGEND + (e * NSEG + seg) * BD + col0 + c] = carry;
}

// ---------------- K3/K5: analytic segment-carry resolve ----------------
__global__ __launch_bounds__(256)
void k_carry(float* __restrict__ ws, int segendOff, int carryOff, int initOff) {
    int tid = blockIdx.x * blockDim.x + threadIdx.x;
    if (tid >= E_DIM * BD) return;
    int e = tid / BD, bd = tid % BD;
    float omaT = ws[WS_OMA + e];
#pragma unroll
    for (int i = 0; i < 9; ++i) omaT *= omaT;   // oma^512
    float cv = ws[initOff + bd];
    for (int seg = 0; seg < NSEG; ++seg) {
        int idx = (e * NSEG + seg) * BD + bd;
        ws[carryOff + idx] = cv;                // value entering this segment
        cv = ws[segendOff + idx] + omaT * cv;
    }
}

// ---------------- K4: final ema (+write) and emv local scan ----------------
__global__ __launch_bounds__(32)
void k_ema_final_emv_local(const float* __restrict__ x, float* __restrict__ out,
                           float* __restrict__ ws) {
    __shared__ __align__(16) float xs[16][PAD];
    __shared__ __align__(16) float es[16][PAD];
    const int lane = threadIdx.x, c = lane & 15, h = lane >> 4;
    const int tile = blockIdx.x, seg = blockIdx.y, e = blockIdx.z;
    const int col0 = tile * 16;
    const float alpha = ws[WS_ALPHA + e], oma = ws[WS_OMA + e];
    v2f a[4], ac;
    build_afrags(alpha, oma, c, h, a, ac);

    float carryE = ws[WS_EMA_CARRY + (e * NSEG + seg) * BD + col0 + c];
    float carryV = 0.f;
    int t0 = seg * SEGLEN;
    for (int ch = 0; ch < NCHUNK; ++ch, t0 += 16) {
        stage_x(x, t0, col0, lane, xs);
        __builtin_prefetch(x + (size_t)(t0 + 16 + (lane & 15)) * BD + col0, 0, 3);
        __syncthreads();
        v2f b[4];
#pragma unroll
        for (int k = 0; k < 4; ++k) {
            int s0 = 4 * k + 2 * h;
            b[k].x = xs[s0][c];
            b[k].y = xs[s0 + 1][c];
        }
        v8f d = scan_chunk(a, ac, b, carryE, h);       // global ema chunk
#pragma unroll
        for (int r = 0; r < 8; ++r) {
            int tl = r + 8 * h;
            out[(size_t)(t0 + tl) * EBD + (col0 + c) * E_DIM + e] = d[r];
            es[tl][c] = d[r];                          // D-layout -> LDS for z
        }
        __syncthreads();
        v2f zb[4];
#pragma unroll
        for (int k = 0; k < 4; ++k) {
            int s0 = 4 * k + 2 * h;
            float z0 = xs[s0][c] - es[s0][c];
            float z1 = xs[s0 + 1][c] - es[s0 + 1][c];
            zb[k].x = z0 * z0;
            zb[k].y = z1 * z1;
        }
        v8f dv = scan_chunk(a, ac, zb, carryV, h);     // local emv chunk
        carryE = __shfl(d[7], 16 + c, 32);
        carryV = __shfl(dv[7], 16 + c, 32);
        __syncthreads();
    }
    if (lane < 16)
        ws[WS_EMV_SEGEND + (e * NSEG + seg) * BD + col0 + c] = carryV;
}

// ---------------- K6: final emv (+write) and nrm ----------------
__global__ __launch_bounds__(32)
void k_emv_final_nrm(const float* __restrict__ x, float* __restrict__ out,
                     const float* __restrict__ ws) {
    __shared__ __align__(16) float xs[16][PAD];
    __shared__ __align__(16) float es[16][PAD];
    const int lane = threadIdx.x, c = lane & 15, h = lane >> 4;
    const int tile = blockIdx.x, seg = blockIdx.y, e = blockIdx.z;
    const int col0 = tile * 16;
    const float alpha = ws[WS_ALPHA + e], oma = ws[WS_OMA + e];
    v2f a[4], ac;
    build_afrags(alpha, oma, c, h, a, ac);

    const float* outE = out;                 // ema plane (written by K4)
    float* outV = out + (size_t)PLANE;
    float* outN = out + (size_t)2 * PLANE;
    float carryV = ws[WS_EMV_CARRY + (e * NSEG + seg) * BD + col0 + c];
    int t0 = seg * SEGLEN;
    for (int ch = 0; ch < NCHUNK; ++ch, t0 += 16) {
        stage_x(x, t0, col0, lane, xs);
#pragma unroll
        for (int i = 0; i < 8; ++i) {        // gather ema (e innermost -> scalar)
            int idx = i * 32 + lane;
            int tt = idx >> 4, cc = idx & 15;
            es[tt][cc] = outE[(size_t)(t0 + tt) * EBD + (col0 + cc) * E_DIM + e];
        }
        __syncthreads();
        v2f zb[4];
#pragma unroll
        for (int k = 0; k < 4; ++k) {
            int s0 = 4 * k + 2 * h;
            float z0 = xs[s0][c] - es[s0][c];
            float z1 = xs[s0 + 1][c] - es[s0 + 1][c];
            zb[k].x = z0 * z0;
            zb[k].y = z1 * z1;
        }
        v8f dv = scan_chunk(a, ac, zb, carryV, h);     // global emv chunk
#pragma unroll
        for (int r = 0; r < 8; ++r) {
            int tl = r + 8 * h;
            size_t addr = (size_t)(t0 + tl) * EBD + (col0 + c) * E_DIM + e;
            float ev = dv[r];
            outV[addr] = ev;
            outN[addr] = (xs[tl][c] - es[tl][c]) * rsqrtf(ev + EPS);
        }
        carryV = __shfl(dv[7], 16 + c, 32);
        __syncthreads();
    }
}

extern "C" void kernel_launch(void* const* d_in, const int* in_sizes, int n_in,
                              void* d_out, int out_size, void* d_ws, size_t ws_size,
                              hipStream_t stream) {
    (void)in_sizes; (void)n_in; (void)out_size; (void)ws_size;
    const float* x    = (const float*)d_in[0];
    const float* logd = (const float*)d_in[1];
    float* out = (float*)d_out;
    float* ws  = (float*)d_ws;

    dim3 g(BD / 16, NSEG, E_DIM);   // 8 x 16 x 16 = 2048 single-wave blocks

    k_init<<<1, 128, 0, stream>>>(x, logd, ws);
    k_ema_local<<<g, 32, 0, stream>>>(x, ws);
    k_carry<<<8, 256, 0, stream>>>(ws, WS_EMA_SEGEND, WS_EMA_CARRY, WS_EMA0);
    k_ema_final_emv_local<<<g, 32, 0, stream>>>(x, out, ws);
    k_carry<<<8, 256, 0, stream>>>(ws, WS_EMV_SEGEND, WS_EMV_CARRY, WS_EMV0);
    k_emv_final_nrm<<<g, 32, 0, stream>>>(x, out, ws);
}